// GraphAttentionLayer_83665962926395
// MI455X (gfx1250) — compile-verified
//
#include <hip/hip_runtime.h>
#include <hip/hip_bf16.h>

#define LRELU_ALPHA 0.2f
#define NEG_BIG (-9.0e15f)

typedef float v2f __attribute__((ext_vector_type(2)));
typedef float v4f __attribute__((ext_vector_type(4)));
typedef float v8f __attribute__((ext_vector_type(8)));
typedef int   v4i __attribute__((ext_vector_type(4)));

__device__ __forceinline__ float lrelu(float x) {
  return (x >= 0.f) ? x : LRELU_ALPHA * x;
}

// ---------------------------------------------------------------------------
// Kernel 1: Wh(1024x64) = h(1024x128) @ W(128x64) via V_WMMA_F32_16X16X4_F32.
// One wave32 per 16x16 output tile; block = 4 waves (one per N-tile).
// ---------------------------------------------------------------------------
__global__ __launch_bounds__(128)
void gat_wh_kernel(const float* __restrict__ h,
                   const float* __restrict__ W,
                   float* __restrict__ Wh) {
  const int lane    = threadIdx.x;        // 0..31
  const int colTile = threadIdx.y;        // 0..3
  const int rowTile = blockIdx.x;         // 0..63
  const int mr = lane & 15;
  const int kb = (lane >> 4) << 1;        // 0 or 2

  v8f acc = {};
  const float* hrow = h + (size_t)(rowTile * 16 + mr) * 128;
  const float* wcol = W + colTile * 16 + mr;

  #pragma unroll 4
  for (int k0 = 0; k0 < 128; k0 += 4) {
    v2f av, bv;
    av.x = hrow[k0 + kb];
    av.y = hrow[k0 + kb + 1];
    bv.x = wcol[(k0 + kb) * 64];
    bv.y = wcol[(k0 + kb + 1) * 64];
    acc = __builtin_amdgcn_wmma_f32_16x16x4_f32(false, av, false, bv,
                                                (short)0, acc, false, false);
  }

  const int mBase = rowTile * 16 + ((lane >> 4) << 3);
  float* out = Wh + colTile * 16 + mr;
  #pragma unroll
  for (int r = 0; r < 8; ++r)
    out[(size_t)(mBase + r) * 64] = acc[r];
}

// ---------------------------------------------------------------------------
// Kernel 2: s1 = Wh @ a[:64], s2 = Wh @ a[64:], two WMMA chains sharing A.
// ---------------------------------------------------------------------------
__global__ __launch_bounds__(128)
void gat_s12_kernel(const float* __restrict__ Wh,
                    const float* __restrict__ a,
                    float* __restrict__ s1,
                    float* __restrict__ s2) {
  const int lane    = threadIdx.x;
  const int colTile = threadIdx.y;
  const int rowTile = blockIdx.x;
  const int mr = lane & 15;
  const int kb = (lane >> 4) << 1;

  const float* a1 = a;
  const float* a2 = a + 64 * 64;

  v8f acc1 = {}, acc2 = {};
  const float* whrow = Wh + (size_t)(rowTile * 16 + mr) * 64;
  const int ncol = colTile * 16 + mr;

  #pragma unroll 4
  for (int k0 = 0; k0 < 64; k0 += 4) {
    v2f av, b1, b2;
    av.x = whrow[k0 + kb];
    av.y = whrow[k0 + kb + 1];
    b1.x = a1[(k0 + kb) * 64 + ncol];
    b1.y = a1[(k0 + kb + 1) * 64 + ncol];
    b2.x = a2[(k0 + kb) * 64 + ncol];
    b2.y = a2[(k0 + kb + 1) * 64 + ncol];
    acc1 = __builtin_amdgcn_wmma_f32_16x16x4_f32(false, av, false, b1,
                                                 (short)0, acc1, false, false);
    acc2 = __builtin_amdgcn_wmma_f32_16x16x4_f32(false, av, false, b2,
                                                 (short)0, acc2, false, false);
  }

  const int mBase = rowTile * 16 + ((lane >> 4) << 3);
  #pragma unroll
  for (int r = 0; r < 8; ++r) {
    s1[(size_t)(mBase + r) * 64 + ncol] = acc1[r];
    s2[(size_t)(mBase + r) * 64 + ncol] = acc2[r];
  }
}

// ---------------------------------------------------------------------------
// Kernel 3: per-row masked softmax over j + fused einsum, float4-vectorized.
// Block = 256 threads = (16 f-quads) x (16 j-groups), one block per row i.
// adj row staged in LDS; attention written with non-temporal b128 stores.
// ---------------------------------------------------------------------------
__global__ __launch_bounds__(256)
void gat_softmax_kernel(const int* __restrict__ adj,
                        const float* __restrict__ Wh,
                        const float* __restrict__ s1,
                        const float* __restrict__ s2,
                        float* __restrict__ att,
                        float* __restrict__ out_h) {
  const int i   = blockIdx.x;
  const int tid = threadIdx.x;
  const int fq  = tid & 15;         // f-quad: covers f = fq*4 .. fq*4+3
  const int jg  = tid >> 4;         // j-group 0..15

  __shared__ int adjs[1024];
  __shared__ v4f redA[16][16];      // [jg][fq]
  __shared__ v4f redB[16][16];
  __shared__ v4f mQ[16];
  __shared__ v4f invQ[16];
  __shared__ v4f numQ[16];

  // Stage adjacency row (4 KB) into LDS: one b128 per thread.
  ((v4i*)adjs)[tid] = ((const v4i*)(adj + (size_t)i * 1024))[tid];

  const v4f* s2q  = (const v4f*)s2;    // (1024,16) of float4
  const v4f* whq  = (const v4f*)Wh;
  const v4f  s1v  = ((const v4f*)s1)[(size_t)i * 16 + fq];

  __syncthreads();

  // ---- Pass 1: masked row max
  v4f mloc;
  #pragma unroll
  for (int c = 0; c < 4; ++c) mloc[c] = NEG_BIG;
  #pragma unroll 4
  for (int j = jg; j < 1024; j += 16) {
    if (adjs[j] > 0) {
      v4f s2v = s2q[(size_t)j * 16 + fq];
      #pragma unroll
      for (int c = 0; c < 4; ++c)
        mloc[c] = fmaxf(mloc[c], lrelu(s1v[c] + s2v[c]));
    }
  }
  redA[jg][fq] = mloc;
  __syncthreads();
  if (tid < 16) {
    v4f m = redA[0][tid];
    #pragma unroll 4
    for (int g = 1; g < 16; ++g) {
      v4f t = redA[g][tid];
      #pragma unroll
      for (int c = 0; c < 4; ++c) m[c] = fmaxf(m[c], t[c]);
    }
    mQ[tid] = m;
  }
  __syncthreads();
  const v4f m = mQ[fq];

  // ---- Pass 2: softmax denominator + fused weighted numerator
  v4f sloc = {}, wloc = {};
  #pragma unroll 4
  for (int j = jg; j < 1024; j += 16) {
    const bool valid = adjs[j] > 0;
    v4f s2v = s2q[(size_t)j * 16 + fq];
    v4f whv = whq[(size_t)j * 16 + fq];
    #pragma unroll
    for (int c = 0; c < 4; ++c) {
      float e = valid ? lrelu(s1v[c] + s2v[c]) : NEG_BIG;
      float p = expf(e - m[c]);    // m==NEG_BIG -> exp(0)=1 -> uniform row
      sloc[c] += p;
      wloc[c] += p * whv[c];
    }
  }
  redA[jg][fq] = sloc;
  redB[jg][fq] = wloc;
  __syncthreads();
  if (tid < 16) {
    v4f d = redA[0][tid], n = redB[0][tid];
    #pragma unroll 4
    for (int g = 1; g < 16; ++g) {
      v4f td = redA[g][tid], tn = redB[g][tid];
      #pragma unroll
      for (int c = 0; c < 4; ++c) { d[c] += td[c]; n[c] += tn[c]; }
    }
    v4f inv;
    #pragma unroll
    for (int c = 0; c < 4; ++c) inv[c] = 1.0f / d[c];
    invQ[tid] = inv;
    numQ[tid] = n;
  }
  __syncthreads();
  const v4f inv = invQ[fq];

  // ---- Pass 3: stream normalized attention (write-once -> non-temporal)
  v4f* attq = (v4f*)(att + (size_t)i * 1024 * 64);
  #pragma unroll 2
  for (int j = jg; j < 1024; j += 16) {
    const bool valid = adjs[j] > 0;
    v4f s2v = s2q[(size_t)j * 16 + fq];
    v4f r;
    #pragma unroll
    for (int c = 0; c < 4; ++c) {
      float e = valid ? lrelu(s1v[c] + s2v[c]) : NEG_BIG;
      r[c] = expf(e - m[c]) * inv[c];
    }
    __builtin_nontemporal_store(r, &attq[(size_t)j * 16 + fq]);
  }

  if (jg == 0) {   // tid 0..15: write elu(h_prime) as one b128 per thread
    v4f hp4 = numQ[fq];
    v4f o;
    #pragma unroll
    for (int c = 0; c < 4; ++c) {
      float hp = hp4[c] * inv[c];
      o[c] = (hp > 0.f) ? hp : expm1f(hp);
    }
    ((v4f*)out_h)[(size_t)i * 16 + fq] = o;
  }
}

// ---------------------------------------------------------------------------
extern "C" void kernel_launch(void* const* d_in, const int* in_sizes, int n_in,
                              void* d_out, int out_size, void* d_ws, size_t ws_size,
                              hipStream_t stream) {
  const float* h   = (const float*)d_in[0];  // (1024,128)
  const int*   adj = (const int*)d_in[1];    // (1024,1024,1)
  const float* W   = (const float*)d_in[2];  // (128,64)
  const float* a   = (const float*)d_in[3];  // (128,64)

  float* Wh = (float*)d_ws;                  // 1024*64
  float* s1 = Wh + 1024 * 64;
  float* s2 = s1 + 1024 * 64;

  float* out_h   = (float*)d_out;            // elu(h_prime): 1024*64
  float* out_att = out_h + 1024 * 64;        // attention: 1024*1024*64

  dim3 blk(32, 4);
  gat_wh_kernel<<<64, blk, 0, stream>>>(h, W, Wh);
  gat_s12_kernel<<<64, blk, 0, stream>>>(Wh, a, s1, s2);
  gat_softmax_kernel<<<1024, 256, 0, stream>>>(adj, Wh, s1, s2, out_att, out_h);
}